// GRACE_33114197852236
// MI455X (gfx1250) — compile-verified
//
#include <hip/hip_runtime.h>
#include <hip/hip_bf16.h>
#include <math.h>

#define N_GENES 3072
#define N_CELLS 3072
#define HIDDEN  64
#define CAUSAL  16
#define CS_DIM  32
#define ALPHA   0.35f
#define NEG_BIG -9.0e15f

// LDS row stride (floats) for [row][k] tiles: even (8B-aligned v2f) and
// 68 = 4 mod 64, so a wave's 32 b64 lanes cover all 64 banks conflict-free.
#define LDSW 68

typedef __attribute__((ext_vector_type(2))) float v2f;
typedef __attribute__((ext_vector_type(8))) float v8f;

// D = A(16x4) * B(4x16) + C(16x16), fp32 exact.
__device__ __forceinline__ v8f wmma_f32(v2f a, v2f b, v8f c) {
  return __builtin_amdgcn_wmma_f32_16x16x4_f32(false, a, false, b, (short)0, c,
                                               false, false);
}

__device__ __forceinline__ v2f lds_v2f(const float* p) {
  return *(const v2f*)p;
}

// ---------------------------------------------------------------------------
// out(M x 64)[m,f] = sum_k X[k*ldx + m0+m] * W[k*64 + f]     (C = X^T @ W)
// grid.x = M/16, block = 128 (4 waves; wave w owns f0 = 16*w).
// A tile shared across waves via LDS; A and B read as ds_load_b64.
// ---------------------------------------------------------------------------
__global__ __launch_bounds__(128) void k_xtw(const float* __restrict__ X,
                                             const float* __restrict__ W,
                                             float* __restrict__ out,
                                             int K, int ldx) {
  __shared__ __attribute__((aligned(16))) float a_lds[16 * LDSW];
  __shared__ __attribute__((aligned(16))) float b_lds[64 * LDSW];
  const int t    = threadIdx.x;
  const int lane = t & 31;
  const int wave = t >> 5;
  const int m    = lane & 15;     // A row / B,C col within tile
  const int half = lane >> 4;     // K pair (A/B), M half (C/D)
  const int m0   = blockIdx.x * 16;
  const int f0   = wave * 16;

  v8f acc = {};
  for (int k0 = 0; k0 < K; k0 += 64) {
    // A: X[k0+kk][m0+mm] -> a_lds[mm][kk]   (1024 elems, coalesced reads)
    for (int it = 0; it < 8; ++it) {
      const int idx = it * 128 + t;
      const int kk = idx >> 4, mm = idx & 15;
      a_lds[mm * LDSW + kk] = X[(size_t)(k0 + kk) * ldx + m0 + mm];
    }
    // B: W[k0+kk][f] -> b_lds[f][kk]        (4096 elems, coalesced reads)
    for (int it = 0; it < 32; ++it) {
      const int idx = it * 128 + t;
      const int kk = idx >> 6, f = idx & 63;
      b_lds[f * LDSW + kk] = W[(size_t)(k0 + kk) * HIDDEN + f];
    }
    __syncthreads();
#pragma unroll
    for (int kk = 0; kk < 64; kk += 4) {
      const int kb = kk + 2 * half;
      acc = wmma_f32(lds_v2f(&a_lds[m * LDSW + kb]),
                     lds_v2f(&b_lds[(f0 + m) * LDSW + kb]), acc);
    }
    __syncthreads();
  }
  for (int v = 0; v < 8; ++v) {
    const int row = v + half * 8;
    out[(size_t)(m0 + row) * HIDDEN + f0 + m] = acc[v];
  }
}

// ---------------------------------------------------------------------------
// s_dst[g] = h[g,:] . a_gat[:64] ; s_src[g] = h[g,:] . a_gat[64:]
// ---------------------------------------------------------------------------
__global__ __launch_bounds__(256) void k_gat_scores(const float* __restrict__ h,
                                                    const float* __restrict__ a_gat,
                                                    float* __restrict__ s_src,
                                                    float* __restrict__ s_dst) {
  const int g = blockIdx.x * blockDim.x + threadIdx.x;
  if (g >= N_GENES) return;
  const float* hr = h + (size_t)g * HIDDEN;
  float sd = 0.f, ss = 0.f;
  for (int f = 0; f < HIDDEN; ++f) {
    sd += hr[f] * a_gat[f];
    ss += hr[f] * a_gat[HIDDEN + f];
  }
  s_dst[g] = sd;
  s_src[g] = ss;
}

// ---------------------------------------------------------------------------
// dis[i] = rsqrt(1 + sum_j adj[i,j])    (degree of adj + I)
// ---------------------------------------------------------------------------
__global__ __launch_bounds__(256) void k_degree(const int* __restrict__ adj,
                                                float* __restrict__ dis) {
  __shared__ float red[256];
  const int i = blockIdx.x;
  const int* row = adj + (size_t)i * N_GENES;
  float s = 0.f;
  for (int j = threadIdx.x; j < N_GENES; j += 256) s += (float)row[j];
  red[threadIdx.x] = s;
  __syncthreads();
  for (int off = 128; off > 0; off >>= 1) {
    if (threadIdx.x < off) red[threadIdx.x] += red[threadIdx.x + off];
    __syncthreads();
  }
  if (threadIdx.x == 0) {
    const float d = red[0] + 1.0f;  // +I
    dis[i] = (d > 0.f) ? rsqrtf(d) : 0.f;
  }
}

// ---------------------------------------------------------------------------
// Masked-softmax GAT row band:  h_gat = elu(softmax(e) @ h), 16 rows/block.
// ---------------------------------------------------------------------------
__global__ __launch_bounds__(128) void k_attention(const int* __restrict__ adj,
                                                   const float* __restrict__ s_src,
                                                   const float* __restrict__ s_dst,
                                                   const float* __restrict__ h,
                                                   float* __restrict__ h_gat) {
  __shared__ float red[128];
  __shared__ float rowmax[16];
  __shared__ float rowsum[16];
  __shared__ __attribute__((aligned(16))) float att[16 * LDSW];
  __shared__ __attribute__((aligned(16))) float h_lds[64 * LDSW];

  const int t    = threadIdx.x;
  const int lane = t & 31;
  const int wave = t >> 5;
  const int m    = lane & 15;
  const int half = lane >> 4;
  const int g0   = blockIdx.x * 16;
  const int f0   = wave * 16;

  const int r = t >> 3;  // row within band, 8 threads per row
  const int q = t & 7;
  const float ssrc = s_src[g0 + r];
  const int* adjrow = adj + (size_t)(g0 + r) * N_GENES;

  // ---- pass 1: row max of masked leaky-relu scores
  float lmax = NEG_BIG;
  for (int j = q; j < N_GENES; j += 8) {
    float e = ssrc + s_dst[j];
    e = (e > 0.f) ? e : ALPHA * e;
    e = (adjrow[j] > 0) ? e : NEG_BIG;
    lmax = fmaxf(lmax, e);
  }
  red[t] = lmax;
  __syncthreads();
  if (t < 16) {
    float mx = red[t * 8];
    for (int u = 1; u < 8; ++u) mx = fmaxf(mx, red[t * 8 + u]);
    rowmax[t] = mx;
  }
  __syncthreads();

  // ---- pass 2: row sum of exp
  const float mx = rowmax[r];
  float lsum = 0.f;
  for (int j = q; j < N_GENES; j += 8) {
    float e = ssrc + s_dst[j];
    e = (e > 0.f) ? e : ALPHA * e;
    e = (adjrow[j] > 0) ? e : NEG_BIG;
    lsum += __expf(e - mx);  // masked -> exact 0 (underflow)
  }
  red[t] = lsum;
  __syncthreads();
  if (t < 16) {
    float s = 0.f;
    for (int u = 0; u < 8; ++u) s += red[t * 8 + u];
    rowsum[t] = s;
  }
  __syncthreads();

  // ---- pass 3: (unnormalized att) @ h via WMMA, K chunks of 64
  v8f acc = {};
  for (int k0 = 0; k0 < N_GENES; k0 += 64) {
    // generate att chunk in LDS
    for (int it = 0; it < 8; ++it) {
      const int idx = it * 128 + t;
      const int rr = idx >> 6, kk = idx & 63;
      const int j  = k0 + kk;
      float e = s_src[g0 + rr] + s_dst[j];
      e = (e > 0.f) ? e : ALPHA * e;
      e = (adj[(size_t)(g0 + rr) * N_GENES + j] > 0) ? e : NEG_BIG;
      att[rr * LDSW + kk] = __expf(e - rowmax[rr]);
    }
    // stage h chunk transposed: h[k0+kk][f] -> h_lds[f][kk]
    for (int it = 0; it < 32; ++it) {
      const int idx = it * 128 + t;
      const int kk = idx >> 6, f = idx & 63;
      h_lds[f * LDSW + kk] = h[(size_t)(k0 + kk) * HIDDEN + f];
    }
    __syncthreads();
#pragma unroll
    for (int kk = 0; kk < 64; kk += 4) {
      const int kb = kk + 2 * half;
      acc = wmma_f32(lds_v2f(&att[m * LDSW + kb]),
                     lds_v2f(&h_lds[(f0 + m) * LDSW + kb]), acc);
    }
    __syncthreads();
  }

  for (int v = 0; v < 8; ++v) {
    const int row = v + half * 8;
    float val = acc[v] / rowsum[row];
    val = (val > 0.f) ? val : expm1f(val);  // ELU
    h_gat[(size_t)(g0 + row) * HIDDEN + f0 + m] = val;
  }
}

// ---------------------------------------------------------------------------
// VAE heads + diffusion fusion: mu, logvar, z, cs per gene.
// ---------------------------------------------------------------------------
__global__ __launch_bounds__(256) void k_vae_diff(
    const float* __restrict__ h_gat, const float* __restrict__ eps,
    const float* __restrict__ W_mu, const float* __restrict__ b_mu,
    const float* __restrict__ W_lv, const float* __restrict__ b_lv,
    const float* __restrict__ W_diff, const float* __restrict__ b_diff,
    float* __restrict__ mu_out, float* __restrict__ lv_out,
    float* __restrict__ cs_out) {
  const int g = blockIdx.x * blockDim.x + threadIdx.x;
  if (g >= N_GENES) return;
  float hv[HIDDEN];
  const float* hr = h_gat + (size_t)g * HIDDEN;
  for (int f = 0; f < HIDDEN; ++f) hv[f] = hr[f];
  float z[CAUSAL];
  for (int c = 0; c < CAUSAL; ++c) {
    float mu = b_mu[c], lv = b_lv[c];
    for (int f = 0; f < HIDDEN; ++f) {
      mu += hv[f] * W_mu[f * CAUSAL + c];
      lv += hv[f] * W_lv[f * CAUSAL + c];
    }
    mu_out[(size_t)g * CAUSAL + c] = mu;
    lv_out[(size_t)g * CAUSAL + c] = lv;
    z[c] = mu + eps[(size_t)g * CAUSAL + c] * __expf(0.5f * lv);
  }
  for (int o = 0; o < CS_DIM; ++o) {
    float a = b_diff[o];
    for (int c = 0; c < CAUSAL; ++c) a += z[c] * W_diff[c * CS_DIM + o];
    for (int f = 0; f < HIDDEN; ++f) a += hv[f] * W_diff[(CAUSAL + f) * CS_DIM + o];
    cs_out[(size_t)g * CS_DIM + o] = fmaxf(a, 0.f);
  }
}

// ---------------------------------------------------------------------------
// x_recon = sigmoid(cs(3072x32) @ W_dec(32x3072) + b_dec)
// grid (48, 192): blockIdx.x -> 64-wide N band, blockIdx.y -> 16-row M band.
// ---------------------------------------------------------------------------
__global__ __launch_bounds__(128) void k_decoder(const float* __restrict__ cs,
                                                 const float* __restrict__ W_dec,
                                                 const float* __restrict__ b_dec,
                                                 float* __restrict__ x_recon) {
  __shared__ __attribute__((aligned(16))) float cst[16 * 34];
  __shared__ __attribute__((aligned(16))) float wd_lds[64 * 34];
  const int t    = threadIdx.x;
  const int lane = t & 31;
  const int wave = t >> 5;
  const int m    = lane & 15;
  const int half = lane >> 4;
  const int g0    = blockIdx.y * 16;
  const int cbase = blockIdx.x * 64;
  const int c0    = cbase + wave * 16;

  for (int idx = t; idx < 16 * 32; idx += 128) {
    const int rr = idx >> 5, kk = idx & 31;
    cst[rr * 34 + kk] = cs[(size_t)(g0 + rr) * CS_DIM + kk];
  }
  for (int idx = t; idx < 64 * 32; idx += 128) {
    const int kk = idx >> 6, cc = idx & 63;
    wd_lds[cc * 34 + kk] = W_dec[(size_t)kk * N_CELLS + cbase + cc];
  }
  __syncthreads();

  v8f acc = {};
#pragma unroll
  for (int k = 0; k < CS_DIM; k += 4) {
    const int kb = k + 2 * half;
    acc = wmma_f32(lds_v2f(&cst[m * 34 + kb]),
                   lds_v2f(&wd_lds[(wave * 16 + m) * 34 + kb]), acc);
  }
  for (int v = 0; v < 8; ++v) {
    const int row = v + half * 8;
    const int c = c0 + m;
    const float val = acc[v] + b_dec[c];
    x_recon[(size_t)(g0 + row) * N_CELLS + c] = 1.0f / (1.0f + __expf(-val));
  }
}

// ---------------------------------------------------------------------------
// out = relu( D^-1/2 (adj+I) D^-1/2 @ T ), T is (3072 x 64).
// A_norm tiles generated on the fly into LDS; T staged transposed.
// ---------------------------------------------------------------------------
__global__ __launch_bounds__(128) void k_anorm_gemm(const int* __restrict__ adj,
                                                    const float* __restrict__ dis,
                                                    const float* __restrict__ T,
                                                    float* __restrict__ out) {
  __shared__ __attribute__((aligned(16))) float a_lds[16 * LDSW];
  __shared__ __attribute__((aligned(16))) float t_lds[64 * LDSW];
  const int t    = threadIdx.x;
  const int lane = t & 31;
  const int wave = t >> 5;
  const int m    = lane & 15;
  const int half = lane >> 4;
  const int i0   = blockIdx.x * 16;
  const int f0   = wave * 16;

  v8f acc = {};
  for (int k0 = 0; k0 < N_GENES; k0 += 64) {
    for (int it = 0; it < 8; ++it) {
      const int idx = it * 128 + t;
      const int rr = idx >> 6, kk = idx & 63;
      const int j = k0 + kk;
      const float av =
          (float)adj[(size_t)(i0 + rr) * N_GENES + j] + ((i0 + rr == j) ? 1.0f : 0.0f);
      a_lds[rr * LDSW + kk] = av * dis[j];
    }
    for (int it = 0; it < 32; ++it) {
      const int idx = it * 128 + t;
      const int kk = idx >> 6, f = idx & 63;
      t_lds[f * LDSW + kk] = T[(size_t)(k0 + kk) * HIDDEN + f];
    }
    __syncthreads();
#pragma unroll
    for (int kk = 0; kk < 64; kk += 4) {
      const int kb = kk + 2 * half;
      acc = wmma_f32(lds_v2f(&a_lds[m * LDSW + kb]),
                     lds_v2f(&t_lds[(f0 + m) * LDSW + kb]), acc);
    }
    __syncthreads();
  }
  for (int v = 0; v < 8; ++v) {
    const int row = v + half * 8;
    const float val = dis[i0 + row] * acc[v];
    out[(size_t)(i0 + row) * HIDDEN + f0 + m] = fmaxf(val, 0.f);
  }
}

// ---------------------------------------------------------------------------
// out(3072x64) = A(3072x64) @ B(64x64)   (A rows contiguous -> global b64)
// ---------------------------------------------------------------------------
__global__ __launch_bounds__(128) void k_small_gemm(const float* __restrict__ A,
                                                    const float* __restrict__ B,
                                                    float* __restrict__ out) {
  const int t    = threadIdx.x;
  const int lane = t & 31;
  const int wave = t >> 5;
  const int m    = lane & 15;
  const int half = lane >> 4;
  const int g0   = blockIdx.x * 16;
  const int f0   = wave * 16;
  v8f acc = {};
#pragma unroll
  for (int k = 0; k < HIDDEN; k += 4) {
    const int kb = k + 2 * half;
    const v2f a = *(const v2f*)&A[(size_t)(g0 + m) * HIDDEN + kb];  // contiguous
    v2f b;
    b.x = B[(size_t)kb       * HIDDEN + f0 + m];
    b.y = B[(size_t)(kb + 1) * HIDDEN + f0 + m];
    acc = wmma_f32(a, b, acc);
  }
  for (int v = 0; v < 8; ++v) {
    const int row = v + half * 8;
    out[(size_t)(g0 + row) * HIDDEN + f0 + m] = acc[v];
  }
}

// ---------------------------------------------------------------------------
// ie = concat([h2 (3072x64), cs (3072x32)], axis=-1) -> (3072x96)
// ---------------------------------------------------------------------------
__global__ __launch_bounds__(256) void k_concat_ie(const float* __restrict__ h2,
                                                   const float* __restrict__ cs,
                                                   float* __restrict__ ie) {
  const int idx = blockIdx.x * blockDim.x + threadIdx.x;
  if (idx >= N_GENES * (HIDDEN + CS_DIM)) return;
  const int g = idx / 96, c = idx % 96;
  ie[idx] = (c < HIDDEN) ? h2[(size_t)g * HIDDEN + c]
                         : cs[(size_t)g * CS_DIM + (c - HIDDEN)];
}

extern "C" void kernel_launch(void* const* d_in, const int* in_sizes, int n_in,
                              void* d_out, int out_size, void* d_ws, size_t ws_size,
                              hipStream_t stream) {
  (void)in_sizes; (void)n_in; (void)out_size; (void)ws_size;
  const float* x      = (const float*)d_in[0];
  const int*   adj    = (const int*)d_in[1];
  const float* eps    = (const float*)d_in[2];
  const float* W_gat  = (const float*)d_in[3];
  const float* a_gat  = (const float*)d_in[4];
  const float* W_mu   = (const float*)d_in[5];
  const float* b_mu   = (const float*)d_in[6];
  const float* W_lv   = (const float*)d_in[7];
  const float* b_lv   = (const float*)d_in[8];
  const float* W_diff = (const float*)d_in[9];
  const float* b_diff = (const float*)d_in[10];
  const float* W_dec  = (const float*)d_in[11];
  const float* b_dec  = (const float*)d_in[12];
  const float* W_gcn1 = (const float*)d_in[13];
  const float* W_gcn2 = (const float*)d_in[14];

  float* out      = (float*)d_out;           // outputs concatenated flat
  float* x_recon  = out;                     // 3072*3072
  float* h2_out   = out + 9437184;           // 3072*64
  float* ie_out   = out + 9633792;           // 3072*96
  float* cs_out   = out + 9928704;           // 3072*32
  float* mu_out   = out + 10027008;          // 3072*16
  float* lv_out   = out + 10076160;          // 3072*16

  float* ws    = (float*)d_ws;               // ~1.6 MB used
  float* buf0  = ws;                         // h, then t1, then t2 (3072*64)
  float* buf1  = ws + 196608;                // h_gat, then h1      (3072*64)
  float* s_src = ws + 393216;                // 3072
  float* s_dst = ws + 396288;                // 3072
  float* dis   = ws + 399360;                // 3072

  // degree normalization (independent of everything but adj)
  k_degree<<<N_GENES, 256, 0, stream>>>(adj, dis);
  // h = x^T @ W_gat
  k_xtw<<<192, 128, 0, stream>>>(x, W_gat, buf0, N_CELLS, N_GENES);
  // attention rank-1 score vectors
  k_gat_scores<<<12, 256, 0, stream>>>(buf0, a_gat, s_src, s_dst);
  // h_gat = elu(softmax(masked e) @ h)
  k_attention<<<192, 128, 0, stream>>>(adj, s_src, s_dst, buf0, buf1);
  // mu / logvar / z / cs
  k_vae_diff<<<12, 256, 0, stream>>>(buf1, eps, W_mu, b_mu, W_lv, b_lv,
                                     W_diff, b_diff, mu_out, lv_out, cs_out);
  // x_recon = sigmoid(cs @ W_dec + b_dec)
  k_decoder<<<dim3(48, 192), 128, 0, stream>>>(cs_out, W_dec, b_dec, x_recon);
  // t1 = x_recon^T @ W_gcn1  (buf0 reused; h is dead)
  k_xtw<<<192, 128, 0, stream>>>(x_recon, W_gcn1, buf0, N_GENES, N_CELLS);
  // h1 = relu(A_norm @ t1)   (buf1 reused; h_gat is dead)
  k_anorm_gemm<<<192, 128, 0, stream>>>(adj, dis, buf0, buf1);
  // t2 = h1 @ W_gcn2         (buf0 reused; t1 is dead)
  k_small_gemm<<<192, 128, 0, stream>>>(buf1, W_gcn2, buf0);
  // h2 = relu(A_norm @ t2)
  k_anorm_gemm<<<192, 128, 0, stream>>>(adj, dis, buf0, h2_out);
  // ie = concat(h2, cs)
  k_concat_ie<<<(N_GENES * 96 + 255) / 256, 256, 0, stream>>>(h2_out, cs_out, ie_out);
}